// FMCTracker_69939247448204
// MI455X (gfx1250) — compile-verified
//
#include <hip/hip_runtime.h>
#include <math.h>

typedef float v2f __attribute__((ext_vector_type(2)));
typedef float v4f __attribute__((ext_vector_type(4)));
typedef float v8f __attribute__((ext_vector_type(8)));

#define NN 512
#define NCHUNK 48
// float offsets inside workspace (all buffers fully written before read)
#define OFF_D    0              // D matrix            (1 MB)
#define OFF_M    (NN*NN)        // M_row, M_col        (2 MB)
#define OFF_T    (3*NN*NN)      // T_row, T_col        (2 MB)
#define OFF_K    (5*NN*NN)      // K_row, K_col        (2 MB)
#define OFF_P    (7*NN*NN)      // partials 64*NCHUNK*2 floats

// ---------------------------------------------------------------------------
// Kernel 0a: ortho DCT-II matrix D[u][j] = c(u)*cos(pi*(2j+1)u / 1024).
// Angle reduced exactly via (2j+1)*u mod 2048 so cosf sees args in [0, 2pi).
// ---------------------------------------------------------------------------
__global__ void k_dctmat(float* __restrict__ Dm) {
    int idx = blockIdx.x * blockDim.x + threadIdx.x;   // 0..262143
    int u = idx >> 9;
    int j = idx & 511;
    int m = ((2 * j + 1) * u) & 2047;
    float ang = (float)m * (float)(M_PI / 1024.0);
    float scale = (u == 0) ? 0.044194173824159216f     // sqrt(1/512)
                           : 0.0625f;                  // sqrt(2/512)
    Dm[idx] = scale * cosf(ang);
}

// ---------------------------------------------------------------------------
// Kernel 0b: weighted attention matrices (removes all branches from the GEMMs)
//   M_row[u,v] = att[u,v]*rw[u],   M_col[u,v] = att[u,v]*cw[v]
// ---------------------------------------------------------------------------
__global__ void k_weight(const float* __restrict__ att, const float* __restrict__ rw,
                         const float* __restrict__ cw, float* __restrict__ M) {
    int idx = blockIdx.x * blockDim.x + threadIdx.x;   // 0..262143
    int u = idx >> 9;
    int v = idx & 511;
    float a = att[idx];
    M[idx]           = a * rw[u];   // M_row
    M[NN * NN + idx] = a * cw[v];   // M_col
}

// ---------------------------------------------------------------------------
// Kernel 1: T_z = M_z @ D. One wave32 per 16x16 tile, fp32 WMMA, K step 4.
// A 16x4: lanes 0-15 K={0,1}, lanes 16-31 K={2,3}, row = lane&15 (b64 load).
// B 4x16 mirrored; C/D: VGPR p -> row p / p+8 split at lane 16, col = lane&15.
// Branch-free inner loop -> loads clause cleanly around v_wmma.
// ---------------------------------------------------------------------------
__global__ void k_gemm_nn(const float* __restrict__ M, const float* __restrict__ Dm,
                          float* __restrict__ T) {
    const int lane = threadIdx.x;        // 0..31
    const int half = lane >> 4;
    const int l16  = lane & 15;
    const int z = blockIdx.z;            // uniform: selects M_row/M_col
    const int rowBase = blockIdx.y << 4;
    const int colBase = blockIdx.x << 4;
    const float* Az = M + z * NN * NN + (rowBase + l16) * NN + half * 2;
    const float* Bz = Dm + half * 2 * NN + colBase + l16;
    v8f acc = {};
#pragma unroll 4
    for (int k = 0; k < NN; k += 4) {
        v2f a = *(const v2f*)(Az + k);               // A[u, k0], A[u, k0+1]
        v2f b; b.x = Bz[k * NN]; b.y = Bz[k * NN + NN];
        acc = __builtin_amdgcn_wmma_f32_16x16x4_f32(
            false, a, false, b, (short)0, acc, false, false);
    }
    float* Tz = T + z * NN * NN;
#pragma unroll
    for (int p = 0; p < 8; ++p)
        Tz[(rowBase + half * 8 + p) * NN + colBase + l16] = acc[p];
}

// ---------------------------------------------------------------------------
// Kernel 2: K_z = D^T @ T_z (A read transposed from D).
// ---------------------------------------------------------------------------
__global__ void k_gemm_tn(const float* __restrict__ Dm, const float* __restrict__ T,
                          float* __restrict__ K) {
    const int lane = threadIdx.x;
    const int half = lane >> 4;
    const int l16  = lane & 15;
    const int z = blockIdx.z;
    const int rowBase = blockIdx.y << 4;   // i
    const int colBase = blockIdx.x << 4;   // j
    const float* Ai = Dm + half * 2 * NN + rowBase + l16;       // D[k0, i]
    const float* Bj = T + z * NN * NN + half * 2 * NN + colBase + l16;
    v8f acc = {};
#pragma unroll 4
    for (int k = 0; k < NN; k += 4) {
        v2f a; a.x = Ai[k * NN]; a.y = Ai[k * NN + NN];
        v2f b; b.x = Bj[k * NN]; b.y = Bj[k * NN + NN];
        acc = __builtin_amdgcn_wmma_f32_16x16x4_f32(
            false, a, false, b, (short)0, acc, false, false);
    }
    float* Kz = K + z * NN * NN;
#pragma unroll
    for (int p = 0; p < 8; ++p)
        Kz[(rowBase + half * 8 + p) * NN + colBase + l16] = acc[p];
}

// ---------------------------------------------------------------------------
// Kernel 3: bandwidth-bound pass (the ~8.6 us roofline floor at 23.3 TB/s).
// Per b: dot(x[b], K_row), dot(x[b], K_col) over 3 channels. x streamed once
// with non-temporal b128 loads; K (2 MB) stays L2-resident. Deterministic
// per-block LDS tree reduction.
// ---------------------------------------------------------------------------
__global__ void k_reduce(const float* __restrict__ x, const float* __restrict__ Kr,
                         const float* __restrict__ Kc, float* __restrict__ partial) {
    const int b  = blockIdx.x;
    const int ch = blockIdx.y;
    const int t  = threadIdx.x;
    const v4f* xb  = (const v4f*)x + (size_t)b * (3 * NN * NN / 4);
    const v4f* Kr4 = (const v4f*)Kr;
    const v4f* Kc4 = (const v4f*)Kc;
    const int per_chunk = (3 * NN * NN / 4) / NCHUNK;   // 196608/48 = 4096
    float ar = 0.f, ac = 0.f;
    for (int q = ch * per_chunk + t; q < (ch + 1) * per_chunk; q += 256) {
        v4f xv = __builtin_nontemporal_load(&xb[q]);
        int pix = q & (NN * NN / 4 - 1);
        v4f kr = Kr4[pix];
        v4f kc = Kc4[pix];
        ar += xv.x * kr.x + xv.y * kr.y + xv.z * kr.z + xv.w * kr.w;
        ac += xv.x * kc.x + xv.y * kc.y + xv.z * kc.z + xv.w * kc.w;
    }
    __shared__ float sr[256];
    __shared__ float sc[256];
    sr[t] = ar; sc[t] = ac;
    __syncthreads();
    for (int s = 128; s > 0; s >>= 1) {
        if (t < s) { sr[t] += sr[t + s]; sc[t] += sc[t + s]; }
        __syncthreads();
    }
    if (t == 0) {
        partial[(b * NCHUNK + ch) * 2 + 0] = sr[0];   // row-weighted
        partial[(b * NCHUNK + ch) * 2 + 1] = sc[0];   // col-weighted
    }
}

// ---------------------------------------------------------------------------
// Kernel 4: deterministic chunk combine, scale, sigmoid.
// out[b,0]=sigmoid(col_out), out[b,1]=sigmoid(row_out)  (stack order).
// ---------------------------------------------------------------------------
__global__ void k_final(const float* __restrict__ partial, float* __restrict__ out) {
    int b = blockIdx.x * blockDim.x + threadIdx.x;
    if (b >= 64) return;
    float sr = 0.f, sc = 0.f;
    for (int ch = 0; ch < NCHUNK; ++ch) {
        sr += partial[(b * NCHUNK + ch) * 2 + 0];
        sc += partial[(b * NCHUNK + ch) * 2 + 1];
    }
    const float scale = 1.0f / (3.0f * 512.0f * 512.0f);
    float ro = sr * scale;
    float co = sc * scale;
    out[b * 2 + 0] = 1.0f / (1.0f + expf(-co));
    out[b * 2 + 1] = 1.0f / (1.0f + expf(-ro));
}

extern "C" void kernel_launch(void* const* d_in, const int* in_sizes, int n_in,
                              void* d_out, int out_size, void* d_ws, size_t ws_size,
                              hipStream_t stream) {
    const float* x   = (const float*)d_in[0];   // [64,3,512,512]
    const float* att = (const float*)d_in[1];   // [512,512]
    const float* rw  = (const float*)d_in[2];   // [512]
    const float* cw  = (const float*)d_in[3];   // [512]
    float* ws = (float*)d_ws;
    float* Dm      = ws + OFF_D;
    float* M       = ws + OFF_M;
    float* T       = ws + OFF_T;
    float* K       = ws + OFF_K;
    float* partial = ws + OFF_P;
    float* out = (float*)d_out;

    k_dctmat <<<dim3(NN * NN / 256), 256, 0, stream>>>(Dm);
    k_weight <<<dim3(NN * NN / 256), 256, 0, stream>>>(att, rw, cw, M);
    k_gemm_nn<<<dim3(32, 32, 2), 32, 0, stream>>>(M, Dm, T);
    k_gemm_tn<<<dim3(32, 32, 2), 32, 0, stream>>>(Dm, T, K);
    k_reduce <<<dim3(64, NCHUNK), 256, 0, stream>>>(x, K, K + NN * NN, partial);
    k_final  <<<dim3(2), 32, 0, stream>>>(partial, out);
}